// Top2Router_77335181132292
// MI455X (gfx1250) — compile-verified
//
#include <hip/hip_runtime.h>
#include <hip/hip_bf16.h>

typedef __attribute__((ext_vector_type(4))) float v4f;

#define NUM_EXPERTS 64

// ---------------------------------------------------------------------------
// Kernel 1: per-token softmax denominator + top1/top2 selection.
// One thread per token; row of 64 f32 loaded as 16x b128 (+ prefetch).
// Matches jnp.argmax first-occurrence semantics (strict >).
// ---------------------------------------------------------------------------
__global__ void router_topk(const float* __restrict__ logits,
                            int* __restrict__ top1, int* __restrict__ top2,
                            float* __restrict__ w1, float* __restrict__ w2,
                            int s) {
  int t = blockIdx.x * blockDim.x + threadIdx.x;
  if (t >= s) return;

  const float* row = logits + (size_t)t * NUM_EXPERTS;
  __builtin_prefetch(row, 0, 1);        // global_prefetch_b8

  float v[NUM_EXPERTS];
#pragma unroll
  for (int i = 0; i < NUM_EXPERTS / 4; ++i) {
    v4f q = *(const v4f*)(row + 4 * i);
    v[4 * i + 0] = q[0];
    v[4 * i + 1] = q[1];
    v[4 * i + 2] = q[2];
    v[4 * i + 3] = q[3];
  }

  // max for softmax stability (== top1 value)
  float m = v[0];
#pragma unroll
  for (int i = 1; i < NUM_EXPERTS; ++i) m = fmaxf(m, v[i]);

  float sum = 0.0f;
#pragma unroll
  for (int i = 0; i < NUM_EXPERTS; ++i) sum += __expf(v[i] - m);

  // top1: first occurrence of max
  float b1 = v[0];
  int i1 = 0;
#pragma unroll
  for (int i = 1; i < NUM_EXPERTS; ++i) {
    if (v[i] > b1) { b1 = v[i]; i1 = i; }
  }
  // top2: first occurrence of max excluding i1
  float b2 = -3.402823466e+38f;
  int i2 = 0;
#pragma unroll
  for (int i = 0; i < NUM_EXPERTS; ++i) {
    if (i != i1 && v[i] > b2) { b2 = v[i]; i2 = i; }
  }

  float inv = 1.0f / sum;
  top1[t] = i1;
  top2[t] = i2;
  w1[t] = __expf(b1 - m) * inv;   // == inv, kept general
  w2[t] = __expf(b2 - m) * inv;
}

// ---------------------------------------------------------------------------
// Kernel 2: per-expert token-ordered exclusive count (the cumsum ranks).
// 64 blocks, one wave32 per expert. Ballot + popc prefix within each chunk
// of 32 tokens keeps token order exact.
// ---------------------------------------------------------------------------
__global__ void expert_scan(const int* __restrict__ top1,
                            const int* __restrict__ top2,
                            int* __restrict__ rank1,
                            int* __restrict__ rank2raw,
                            int* __restrict__ cnt1_total,
                            int s) {
  const int e = blockIdx.x;
  const int lane = threadIdx.x;              // 0..31 (wave32)
  const unsigned lt_mask = (1u << lane) - 1u;

  int c1 = 0, c2 = 0;
  for (int base = 0; base < s; base += 32) {
    int t = base + lane;
    bool in_range = (t < s);
    bool is1 = in_range && (top1[t] == e);
    bool is2 = in_range && (top2[t] == e);
    unsigned m1 = (unsigned)__ballot(is1);   // wave32: low 32 bits
    unsigned m2 = (unsigned)__ballot(is2);
    if (is1) rank1[t] = c1 + __popc(m1 & lt_mask);
    if (is2) rank2raw[t] = c2 + __popc(m2 & lt_mask);
    c1 += __popc(m1);
    c2 += __popc(m2);
  }
  if (lane == 0) cnt1_total[e] = c1;
}

// ---------------------------------------------------------------------------
// Kernel 3a: zero-fill via the CDNA5 async LDS->global store engine.
// A 256 B zero tile lives in LDS; each lane issues a non-temporal
// global_store_async_from_lds_b128 (512 B per wave per issue, ASYNCcnt
// tracked), drained with s_wait_asynccnt 0. NT keeps the 0.5 GB zero
// stream from displacing lines in the 192 MB L2.
// ---------------------------------------------------------------------------
__global__ void zero_fill_async(float* __restrict__ out, size_t n16) {
  __shared__ __align__(16) float zbuf[64];   // 256 B of zeros
  if (threadIdx.x < 64) zbuf[threadIdx.x] = 0.0f;
  __syncthreads();

  // Low 32 bits of a generic pointer to LDS == LDS byte offset (VSRC field).
  // Spread lanes over 16 distinct 16B tiles to spread LDS bank reads.
  unsigned lds_word = (unsigned)(unsigned long long)(&zbuf[0])
                    + ((threadIdx.x & 15u) * 16u);

  size_t i = (size_t)blockIdx.x * blockDim.x + threadIdx.x;
  const size_t stride = (size_t)gridDim.x * blockDim.x;
  for (size_t k = i; k < n16; k += stride) {
    unsigned long long ga = (unsigned long long)(out + 4 * k);
    asm volatile("global_store_async_from_lds_b128 %0, %1, off th:TH_STORE_NT"
                 :
                 : "v"(ga), "v"(lds_word)
                 : "memory");
  }
  asm volatile("s_wait_asynccnt 0x0" ::: "memory");
}

// ---------------------------------------------------------------------------
// Kernel 3b: zero-fill with non-temporal b128 stores (mask half).
// ---------------------------------------------------------------------------
__global__ void zero_fill_nt(float* __restrict__ out, size_t n4, size_t n_total) {
  size_t i = (size_t)blockIdx.x * blockDim.x + threadIdx.x;
  const size_t stride = (size_t)gridDim.x * blockDim.x;
  v4f z = (v4f)(0.0f);
  v4f* p = (v4f*)out;
  for (size_t k = i; k < n4; k += stride) {
    __builtin_nontemporal_store(z, p + k);
  }
  if (i == 0) {  // scalar tail (none for this shape, kept for safety)
    for (size_t r = n4 * 4; r < n_total; ++r) out[r] = 0.0f;
  }
}

// ---------------------------------------------------------------------------
// Kernel 4: scatter the <=2 nonzero (weight, mask) pairs per token.
// ---------------------------------------------------------------------------
__global__ void scatter_topk(const int* __restrict__ top1,
                             const int* __restrict__ top2,
                             const float* __restrict__ w1,
                             const float* __restrict__ w2,
                             const int* __restrict__ rank1,
                             const int* __restrict__ rank2raw,
                             const int* __restrict__ cnt1_total,
                             float* __restrict__ out,
                             int s, int capacity) {
  int t = blockIdx.x * blockDim.x + threadIdx.x;
  if (t >= s) return;

  const size_t cap = (size_t)capacity;
  const size_t base = (size_t)t * NUM_EXPERTS * cap;
  const size_t mask_off = (size_t)s * NUM_EXPERTS * cap;

  int e1 = top1[t];
  int r1 = rank1[t];
  if (r1 < capacity) {
    size_t idx = base + (size_t)e1 * cap + (size_t)r1;
    out[idx] = w1[t];
    out[mask_off + idx] = 1.0f;   // sec_mask (bool) as 1.0
  }

  int e2 = top2[t];
  int r2 = rank2raw[t] + cnt1_total[e2];
  if (r2 < capacity) {
    size_t idx = base + (size_t)e2 * cap + (size_t)r2;
    out[idx] = w2[t];
    out[mask_off + idx] = 1.0f;
  }
}

// ---------------------------------------------------------------------------
extern "C" void kernel_launch(void* const* d_in, const int* in_sizes, int n_in,
                              void* d_out, int out_size, void* d_ws, size_t ws_size,
                              hipStream_t stream) {
  const float* logits = (const float*)d_in[0];
  const int s = in_sizes[0] / NUM_EXPERTS;           // 8192

  // capacity: floor(2.0 * s / e), round up to even, min 4  -> 256 here
  int capacity = (int)((2.0 * (double)s) / (double)NUM_EXPERTS);
  capacity += capacity % 2;
  if (capacity < 4) capacity = 4;

  float* out = (float*)d_out;

  // workspace layout
  int*   top1     = (int*)d_ws;
  int*   top2     = top1 + s;
  float* w1       = (float*)(top2 + s);
  float* w2       = w1 + s;
  int*   rank1    = (int*)(w2 + s);
  int*   rank2raw = rank1 + s;
  int*   cnt1     = rank2raw + s;

  const int tpb = 256;
  router_topk<<<(s + tpb - 1) / tpb, tpb, 0, stream>>>(logits, top1, top2, w1, w2, s);

  expert_scan<<<NUM_EXPERTS, 32, 0, stream>>>(top1, top2, rank1, rank2raw, cnt1, s);

  const size_t n_total = (size_t)out_size;           // 2 * s * e * c floats
  const size_t n_half = n_total / 2;                 // cb_weight half (16B aligned)
  const size_t n16_w  = n_half / 4;                  // 16B chunks in weight half

  // cb_weight half: async LDS->global store engine (non-temporal)
  zero_fill_async<<<4096, tpb, 0, stream>>>(out, n16_w);
  // sec_mask half (+ any tail): non-temporal b128 stores
  zero_fill_nt<<<4096, tpb, 0, stream>>>(out + n_half, (n_total - n_half) / 4, n_total - n_half);

  scatter_topk<<<(s + tpb - 1) / tpb, tpb, 0, stream>>>(top1, top2, w1, w2,
                                                        rank1, rank2raw, cnt1,
                                                        out, s, capacity);
}